// OscillatorySSMLayer_1460288881149
// MI455X (gfx1250) — compile-verified
//
#include <hip/hip_runtime.h>
#include <math.h>

// Problem constants (match reference setup_inputs).
#define BB 4
#define SS 2048
#define DD 1024
#define NNh 64
#define MM (BB * SS)      // 8192 rows
#define KX (2 * DD)       // 2048 (interleaved re/im feature dim of x rows)
#define NP 192            // proj cols: [Bxr(64) | Bxi(64) | gate(64)]
#define KH (2 * NNh)      // 128 (interleaved h_all row)
#define NO (2 * DD)       // 2048 output cols (interleaved re/im)

#define A2LD 144          // A tile pair-row stride in float2 units (128 + 16 pad)

typedef __attribute__((ext_vector_type(2))) float v2f;
typedef __attribute__((ext_vector_type(8))) float v8f;

__device__ __forceinline__ v8f wmma_f32(v2f a, v2f b, v8f c) {
  // D = A(16x4 f32) * B(4x16 f32) + C(16x16 f32)
  return __builtin_amdgcn_wmma_f32_16x16x4_f32(
      /*neg_a=*/false, a, /*neg_b=*/false, b,
      /*c_mod=*/(short)0, c, /*reuse_a=*/false, /*reuse_b=*/false);
}

// ---------------------------------------------------------------------------
// Tiled fp32 WMMA GEMM core. Block tile: 128(M) x (NWN*32)(N), 8 wave32 waves
// in a 4(M) x 2(N) grid; each wave owns a 32 x (NWN*16) tile = 2 x NWN WMMA
// accumulators. LDS tiles are K-pair-major float2 so every WMMA fragment is a
// single aligned ds_load_b64 straight into an even VGPR pair.
// ---------------------------------------------------------------------------
template <int NWN>
__device__ void gemm_tile_acc(const float* __restrict__ A, int lda,
                              const float* __restrict__ Bp, int ldb,
                              int K, int m0, int j0,
                              v2f* As2, v2f* Bs2, v8f acc[2][NWN]) {
  constexpr int BN   = NWN * 32;      // block N extent
  constexpr int B2LD = BN + 16;       // B tile pair-row stride (float2 units)
  constexpr int NB4  = BN / 64;       // float4 loads of B per thread per stage

  const int tid  = threadIdx.x;
  const int lane = tid & 31;
  const int wave = tid >> 5;
  const int half = lane >> 4;          // 0: lanes 0-15, 1: lanes 16-31
  const int lrow = lane & 15;
  const int wm   = (wave & 3) * 32;    // wave M offset in tile
  const int wn   = (wave >> 2) * (NWN * 16);  // wave N offset in tile

  // Global->LDS staging assignment.
  // A tile: 128 rows x 16 k, 2 float4 per thread (k-contiguous => coalesced).
  const int am0 = tid >> 2;
  const int ak0 = (tid & 3) << 2;
  const int am1 = (256 + tid) >> 2;
  const int ak1 = ((256 + tid) & 3) << 2;
  // B tile: 16 k x BN cols, NB4 float4 per thread.
  int bk[NB4], bj[NB4];
#pragma unroll
  for (int i = 0; i < NB4; ++i) {
    int idx = i * 256 + tid;
    bk[i] = idx / (BN / 4);
    bj[i] = (idx % (BN / 4)) * 4;
  }

  float4 aReg0 = *(const float4*)(A + (size_t)(m0 + am0) * lda + ak0);
  float4 aReg1 = *(const float4*)(A + (size_t)(m0 + am1) * lda + ak1);
  float4 bReg[NB4];
#pragma unroll
  for (int i = 0; i < NB4; ++i)
    bReg[i] = *(const float4*)(Bp + (size_t)bk[i] * ldb + j0 + bj[i]);

  for (int k0 = 0; k0 < K; k0 += 16) {
    __syncthreads();
    // A store, K-pair-major: As2[k/2][m] = {A[m][k], A[m][k+1]}
    {
      v2f t;
      t.x = aReg0.x; t.y = aReg0.y; As2[(ak0 >> 1) * A2LD + am0] = t;
      t.x = aReg0.z; t.y = aReg0.w; As2[((ak0 >> 1) + 1) * A2LD + am0] = t;
      t.x = aReg1.x; t.y = aReg1.y; As2[(ak1 >> 1) * A2LD + am1] = t;
      t.x = aReg1.z; t.y = aReg1.w; As2[((ak1 >> 1) + 1) * A2LD + am1] = t;
    }
    // B store, K-pair transpose: Bs2[k/2][j].(k&1) = B[k][j]
    {
      float* bsf = (float*)Bs2;
#pragma unroll
      for (int i = 0; i < NB4; ++i) {
        int base = ((bk[i] >> 1) * B2LD + bj[i]) * 2 + (bk[i] & 1);
        bsf[base + 0] = bReg[i].x;
        bsf[base + 2] = bReg[i].y;
        bsf[base + 4] = bReg[i].z;
        bsf[base + 6] = bReg[i].w;
      }
    }
    __syncthreads();

    if (k0 + 16 < K) {  // register prefetch of next stage overlaps WMMA math
      aReg0 = *(const float4*)(A + (size_t)(m0 + am0) * lda + (k0 + 16) + ak0);
      aReg1 = *(const float4*)(A + (size_t)(m0 + am1) * lda + (k0 + 16) + ak1);
#pragma unroll
      for (int i = 0; i < NB4; ++i)
        bReg[i] = *(const float4*)(Bp + (size_t)(k0 + 16 + bk[i]) * ldb + j0 + bj[i]);
    }
    if (k0 + 32 < K) {  // L2/L0 prefetch two stages ahead (global_prefetch_b8)
      __builtin_prefetch(A + (size_t)(m0 + am0) * lda + (k0 + 32) + ak0, 0, 0);
      __builtin_prefetch(A + (size_t)(m0 + am1) * lda + (k0 + 32) + ak1, 0, 0);
    }

#pragma unroll
    for (int kstep = 0; kstep < 4; ++kstep) {
      // A-fragment: lanes 0-15 need K={4k,4k+1} (pair row 2k), lanes 16-31
      // need K={4k+2,4k+3} (pair row 2k+1) -> p = 2*kstep + half.
      const int p = 2 * kstep + half;
      v2f a0 = As2[p * A2LD + wm + lrow];
      v2f a1 = As2[p * A2LD + wm + 16 + lrow];
      v2f bf[NWN];
#pragma unroll
      for (int ni = 0; ni < NWN; ++ni)
        bf[ni] = Bs2[p * B2LD + wn + ni * 16 + lrow];
#pragma unroll
      for (int ni = 0; ni < NWN; ++ni) {
        acc[0][ni] = wmma_f32(a0, bf[ni], acc[0][ni]);
        acc[1][ni] = wmma_f32(a1, bf[ni], acc[1][ni]);
      }
    }
  }
}

// ---------------------------------------------------------------------------
// Weight packing kernels (once per launch; packed weights stay L2-resident).
// ---------------------------------------------------------------------------
__global__ void pack_w1_kernel(const float* __restrict__ Bwr, const float* __restrict__ Bwi,
                               const float* __restrict__ gw, float* __restrict__ W1) {
  int idx = blockIdx.x * 256 + threadIdx.x;
  if (idx >= KX * NP) return;
  int k = idx / NP, j = idx - k * NP;
  int d = k >> 1, p = k & 1;
  float v;
  if (j < 64) {                 // yr:  xr*Bwr - xi*Bwi
    v = p ? -Bwi[j * DD + d] : Bwr[j * DD + d];
  } else if (j < 128) {         // yi:  xr*Bwi + xi*Bwr
    int n = j - 64;
    v = p ? Bwr[n * DD + d] : Bwi[n * DD + d];
  } else {                      // gate logits: [xr|xi] @ gate_w.T
    int n = j - 128;
    v = gw[n * (2 * DD) + p * DD + d];
  }
  W1[idx] = v;
}

__global__ void pack_wd_kernel(const float* __restrict__ Dr, const float* __restrict__ Di,
                               float* __restrict__ WD) {
  int idx = blockIdx.x * 256 + threadIdx.x;
  if (idx >= KX * NO) return;
  int k = idx >> 11, j = idx & (NO - 1);
  int e = k >> 1, p = k & 1, d = j >> 1;
  int q = j & 1;
  float wr = Dr[d * DD + e], wi = Di[d * DD + e];
  WD[idx] = (p == q) ? wr : (p ? -wi : wi);
}

__global__ void pack_wc_kernel(const float* __restrict__ Cr, const float* __restrict__ Ci,
                               float* __restrict__ WC) {
  int idx = blockIdx.x * 256 + threadIdx.x;
  if (idx >= KH * NO) return;
  int k = idx >> 11, j = idx & (NO - 1);
  int n = k >> 1, p = k & 1, d = j >> 1;
  int q = j & 1;
  float wr = Cr[d * NNh + n], wi = Ci[d * NNh + n];
  WC[idx] = (p == q) ? wr : (p ? -wi : wi);
}

// ---------------------------------------------------------------------------
// Kernel 1: proj GEMM  P[8192 x 192] = x[8192 x 2048] @ W1, sigmoid on gates.
// ---------------------------------------------------------------------------
__global__ __launch_bounds__(256) void gemm_proj_kernel(const float* __restrict__ x,
                                                        const float* __restrict__ W1,
                                                        const float* __restrict__ gate_b,
                                                        float* __restrict__ P) {
  __shared__ v2f As2[8 * A2LD];
  __shared__ v2f Bs2[8 * (64 + 16)];
  v8f acc[2][2] = {{{0,0,0,0,0,0,0,0}, {0,0,0,0,0,0,0,0}},
                   {{0,0,0,0,0,0,0,0}, {0,0,0,0,0,0,0,0}}};
  const int m0 = blockIdx.x * 128;
  const int j0 = blockIdx.y * 64;
  gemm_tile_acc<2>(x, KX, W1, NP, KX, m0, j0, As2, Bs2, acc);

  const int lane = threadIdx.x & 31, wave = threadIdx.x >> 5;
  const int half = lane >> 4, lrow = lane & 15;
  const int wm = (wave & 3) * 32, wn = (wave >> 2) * 32;
#pragma unroll
  for (int mi = 0; mi < 2; ++mi)
#pragma unroll
    for (int ni = 0; ni < 2; ++ni)
#pragma unroll
      for (int e = 0; e < 8; ++e) {
        int m = m0 + wm + mi * 16 + half * 8 + e;
        int j = j0 + wn + ni * 16 + lrow;
        float v = acc[mi][ni][e];
        if (j >= 128) v = 1.0f / (1.0f + expf(-(v + gate_b[j - 128])));
        P[(size_t)m * NP + j] = v;
      }
}

// ---------------------------------------------------------------------------
// Kernel 2: recurrent scan. 256 threads = 256 independent (b,n) chains.
// ---------------------------------------------------------------------------
__global__ __launch_bounds__(256) void scan_kernel(const float* __restrict__ P,
                                                   const float* __restrict__ h0,
                                                   const float* __restrict__ theta,
                                                   const float* __restrict__ dampp,
                                                   float* __restrict__ hws,
                                                   float* __restrict__ hfin) {
  const int t = threadIdx.x;
  const int b = t >> 6, n = t & 63;
  const float c = cosf(theta[n]);
  const float s = sinf(theta[n]);
  const float dmp = 0.5f + 0.5f / (1.0f + expf(-dampp[n]));
  const float ca = c * dmp, sa = s * dmp;  // fold damping into rotation
  float hr = h0[(b * NNh + n) * 2 + 0];
  float hi = h0[(b * NNh + n) * 2 + 1];
#pragma unroll 4
  for (int st = 0; st < SS; ++st) {
    size_t m = (size_t)b * SS + st;
    float g  = P[m * NP + 128 + n];
    float br = P[m * NP + n];
    float bi = P[m * NP + 64 + n];
    float rr = hr * ca - hi * sa;
    float ri = hr * sa + hi * ca;
    hr = g * br + (1.0f - g) * rr;
    hi = g * bi + (1.0f - g) * ri;
    hws[m * KH + 2 * n + 0] = hr;
    hws[m * KH + 2 * n + 1] = hi;
  }
  hfin[(b * NNh + n) * 2 + 0] = hr;
  hfin[(b * NNh + n) * 2 + 1] = hi;
}

// ---------------------------------------------------------------------------
// Kernel 3: output GEMM  z = x @ WD + h_all @ WC, then out = z_r * z + z.
// z_r lives in the even-lane partner of the C fragment -> wave32 shuffle.
// ---------------------------------------------------------------------------
__global__ __launch_bounds__(256) void gemm_out_kernel(const float* __restrict__ x,
                                                       const float* __restrict__ WD,
                                                       const float* __restrict__ hws,
                                                       const float* __restrict__ WC,
                                                       float* __restrict__ out) {
  __shared__ v2f As2[8 * A2LD];
  __shared__ v2f Bs2[8 * (128 + 16)];
  v8f acc[2][4] = {{{0,0,0,0,0,0,0,0}, {0,0,0,0,0,0,0,0},
                    {0,0,0,0,0,0,0,0}, {0,0,0,0,0,0,0,0}},
                   {{0,0,0,0,0,0,0,0}, {0,0,0,0,0,0,0,0},
                    {0,0,0,0,0,0,0,0}, {0,0,0,0,0,0,0,0}}};
  const int m0 = blockIdx.x * 128;
  const int j0 = blockIdx.y * 128;
  gemm_tile_acc<4>(x, KX, WD, NO, KX, m0, j0, As2, Bs2, acc);    // phase 1: Dx
  __syncthreads();
  gemm_tile_acc<4>(hws, KH, WC, NO, KH, m0, j0, As2, Bs2, acc);  // phase 2: C @ h_all

  const int lane = threadIdx.x & 31, wave = threadIdx.x >> 5;
  const int half = lane >> 4, lrow = lane & 15;
  const int wm = (wave & 3) * 32, wn = (wave >> 2) * 64;
#pragma unroll
  for (int mi = 0; mi < 2; ++mi)
#pragma unroll
    for (int ni = 0; ni < 4; ++ni)
#pragma unroll
      for (int e = 0; e < 8; ++e) {
        int m = m0 + wm + mi * 16 + half * 8 + e;
        int j = j0 + wn + ni * 16 + lrow;
        float z = acc[mi][ni][e];
        float zr = __shfl(z, lane & ~1, 32);  // real component of this (m, d) pair
        out[(size_t)m * NO + j] = zr * z + z;
      }
}

// ---------------------------------------------------------------------------
// Launch. Inputs (setup_inputs order): x, h, theta, damping_param,
// B_wr, B_wi, C_wr, C_wi, D_wr, D_wi, gate_w, gate_b.
// d_out = out[4,2048,1024,2] (16,777,216 f32) ++ h_final[4,64,2] (512 f32).
// Workspace (~29.9 MB): W1 | WD | WC | P | h_all.
// ---------------------------------------------------------------------------
extern "C" void kernel_launch(void* const* d_in, const int* in_sizes, int n_in,
                              void* d_out, int out_size, void* d_ws, size_t ws_size,
                              hipStream_t stream) {
  const float* x     = (const float*)d_in[0];
  const float* h0    = (const float*)d_in[1];
  const float* th    = (const float*)d_in[2];
  const float* dampp = (const float*)d_in[3];
  const float* Bwr   = (const float*)d_in[4];
  const float* Bwi   = (const float*)d_in[5];
  const float* Cwr   = (const float*)d_in[6];
  const float* Cwi   = (const float*)d_in[7];
  const float* Dwr   = (const float*)d_in[8];
  const float* Dwi   = (const float*)d_in[9];
  const float* gw    = (const float*)d_in[10];
  const float* gb    = (const float*)d_in[11];

  float* out  = (float*)d_out;
  float* hfin = out + (size_t)MM * NO;

  float* ws = (float*)d_ws;
  float* W1  = ws;                         // 2048*192
  float* WD  = W1 + (size_t)KX * NP;       // 2048*2048
  float* WC  = WD + (size_t)KX * NO;       // 128*2048
  float* P   = WC + (size_t)KH * NO;       // 8192*192
  float* Hws = P + (size_t)MM * NP;        // 8192*128

  pack_w1_kernel<<<(KX * NP + 255) / 256, 256, 0, stream>>>(Bwr, Bwi, gw, W1);
  pack_wd_kernel<<<(KX * NO + 255) / 256, 256, 0, stream>>>(Dwr, Dwi, WD);
  pack_wc_kernel<<<(KH * NO + 255) / 256, 256, 0, stream>>>(Cwr, Cwi, WC);

  gemm_proj_kernel<<<dim3(MM / 128, NP / 64), 256, 0, stream>>>(x, W1, gb, P);
  scan_kernel<<<1, 256, 0, stream>>>(P, h0, th, dampp, Hws, hfin);
  gemm_out_kernel<<<dim3(MM / 128, NO / 128), 256, 0, stream>>>(x, WD, Hws, WC, out);
}